// DeepseekAttention_44710609551709
// MI455X (gfx1250) — compile-verified
//
#include <hip/hip_runtime.h>
#include <hip/hip_bf16.h>
#include <math.h>

typedef float v2f __attribute__((ext_vector_type(2)));
typedef float v8f __attribute__((ext_vector_type(8)));
typedef unsigned int u32x4 __attribute__((ext_vector_type(4)));
typedef int i32x4 __attribute__((ext_vector_type(4)));
typedef int i32x8 __attribute__((ext_vector_type(8)));

#define BATCH    32
#define HIDDEN   5120
#define NHEADS   128
#define KVLORA   512
#define QLORA    1536
#define ROPED    64
#define NOPED    128
#define VD       128
#define QHD      192
#define KVLEN    4096
#define KVDIM    576                     // KV_LORA + ROPE_D
#define SM_SCALE 0.07216878364870323f    // 1/sqrt(192)

#if defined(__AMDGCN__) && __has_builtin(__builtin_amdgcn_tensor_load_to_lds) && \
    __has_builtin(__builtin_amdgcn_s_wait_tensorcnt)
#define HAVE_TDM 1
#else
#define HAVE_TDM 0
#endif

// ---------------------------------------------------------------------------
// fp32 WMMA primitive: D(16x16) = A(16x4) * B(4x16) + C
// A frag: lane(l) -> row l&15, K-pair (l>>4)*2        (2 VGPRs / lane)
// B frag: lane(l) -> col l&15, K-pair (l>>4)*2
// C/D   : VGPR t of lane l -> row t + 8*(l>>4), col l&15
// ---------------------------------------------------------------------------
static __device__ __forceinline__ v8f wmma4(v2f a, v2f b, v8f c) {
  return __builtin_amdgcn_wmma_f32_16x16x4_f32(false, a, false, b, (short)0, c,
                                               false, false);
}

// 16x16 C-tile of  A(MxK row-major, lda) @ W(NxK row-major, ldw)^T
static __device__ __forceinline__ v8f tile_AxWt(const float* __restrict__ A, int lda,
                                                const float* __restrict__ W, int ldw,
                                                int m0, int n0, int K, int lane) {
  const int r  = lane & 15;
  const int hh = (lane >> 4) << 1;
  const float* ap = A + (size_t)(m0 + r) * lda + hh;
  const float* wp = W + (size_t)(n0 + r) * ldw + hh;
  v8f c = {0.f, 0.f, 0.f, 0.f, 0.f, 0.f, 0.f, 0.f};
#pragma unroll 4
  for (int k = 0; k < K; k += 4) {
    v2f a = {ap[k], ap[k + 1]};
    v2f b = {wp[k], wp[k + 1]};
    c = wmma4(a, b, c);
  }
  return c;
}

// 16x16 C-tile of  A(MxK row-major, lda) @ B(KxN row-major, ldb)
static __device__ __forceinline__ v8f tile_AxB(const float* __restrict__ A, int lda,
                                               const float* __restrict__ B, int ldb,
                                               int m0, int n0, int K, int lane) {
  const int r  = lane & 15;
  const int hh = (lane >> 4) << 1;
  const float* ap = A + (size_t)(m0 + r) * lda + hh;
  const float* bp = B + (size_t)hh * ldb + n0 + r;
  v8f c = {0.f, 0.f, 0.f, 0.f, 0.f, 0.f, 0.f, 0.f};
#pragma unroll 4
  for (int k = 0; k < K; k += 4) {
    v2f a = {ap[k], ap[k + 1]};
    v2f b = {bp[(size_t)k * ldb], bp[(size_t)(k + 1) * ldb]};
    c = wmma4(a, b, c);
  }
  return c;
}

static __device__ __forceinline__ void store_tile(float* __restrict__ C, int ldc,
                                                  int m0, int n0, v8f c, int lane,
                                                  float scale) {
  const int col = lane & 15;
  const int h8  = (lane >> 4) * 8;
#pragma unroll
  for (int t = 0; t < 8; ++t)
    C[(size_t)(m0 + t + h8) * ldc + n0 + col] = c[t] * scale;
}

// ---------------------------------------------------------------------------
// TDM: DMA one 16-row x 576-col fp32 tile (row stride 576 elems in global)
// into LDS at byte offset `lds_off`, with 1 dword of padding inserted after
// every 256 dwords (pad_interval code 7, pad_amount code 0) so LDS element L
// lives at dword (L + (L>>8)) -> bank-conflict-free row access, and b64
// pairs (even L) never straddle a pad.
// ---------------------------------------------------------------------------
#define TILE_DW 9216                      // 16 * 576
#define BUF_DW  9252                      // + 36 pad dwords
static __device__ __forceinline__ void tdm_load_tile(unsigned lds_off,
                                                     const float* gsrc) {
#if HAVE_TDM
  const unsigned long long ga = (unsigned long long)(size_t)gsrc;
  u32x4 g0;
  g0.x = 1u;                                    // count=1, user descriptor
  g0.y = lds_off;                               // LDS dest address (bytes)
  g0.z = (unsigned)ga;                          // global addr [31:0]
  g0.w = (unsigned)(ga >> 32) | (2u << 30);     // addr [56:32] | type=2 (image)
  i32x8 g1;
  g1[0] = (2 << 16) | (1 << 20) | (7 << 22);    // 4B elems, pad 1dw / 256dw
  g1[1] = (int)(576u << 16);                    // tensor_dim0 = 576
  g1[2] = (int)(4096u << 16);                   // tensor_dim1 = 4096
  g1[3] = (int)(576u << 16);                    // tile_dim0 = 576
  g1[4] = 16;                                   // tile_dim1 = 16, tile_dim2 = 0
  g1[5] = 576;                                  // tensor_dim0_stride = 576
  g1[6] = 0;
  g1[7] = 0;
  const i32x4 z4 = {0, 0, 0, 0};
#if __clang_major__ >= 23
  const i32x8 z8 = {0, 0, 0, 0, 0, 0, 0, 0};
  __builtin_amdgcn_tensor_load_to_lds(g0, g1, z4, z4, z8, 0);
#else
  __builtin_amdgcn_tensor_load_to_lds(g0, g1, z4, z4, 0);
#endif
#else
  (void)lds_off; (void)gsrc;
#endif
}

// ---------------------------------------------------------------------------
// Generic M=32 GEMM:  C(32xN) = A(32xK) @ W(NxK)^T, 64 cols per block, 4 waves
// ---------------------------------------------------------------------------
__global__ __launch_bounds__(128) void k_gemm32_AxWt(const float* __restrict__ A, int lda,
                                                     const float* __restrict__ W, int ldw,
                                                     float* __restrict__ C, int ldc, int K) {
  const int lane = threadIdx.x & 31;
  const int wave = threadIdx.x >> 5;
  const int n0 = blockIdx.x * 64 + wave * 16;
#pragma unroll
  for (int mt = 0; mt < 2; ++mt) {
    v8f c = tile_AxWt(A, lda, W, ldw, mt * 16, n0, K, lane);
    store_tile(C, ldc, mt * 16, n0, c, lane, 1.0f);
  }
}

// ---------------------------------------------------------------------------
// RMS norm
// ---------------------------------------------------------------------------
__global__ __launch_bounds__(256) void k_rms(const float* __restrict__ q_a,
                                             const float* __restrict__ lnw,
                                             float* __restrict__ q_norm) {
  const int b = blockIdx.x;
  __shared__ float red[256];
  float s = 0.f;
  for (int k = threadIdx.x; k < QLORA; k += 256) {
    float v = q_a[(size_t)b * QLORA + k];
    s += v * v;
  }
  red[threadIdx.x] = s;
  __syncthreads();
  for (int o = 128; o > 0; o >>= 1) {
    if (threadIdx.x < o) red[threadIdx.x] += red[threadIdx.x + o];
    __syncthreads();
  }
  const float inv = rsqrtf(red[0] / (float)QLORA + 1e-6f);
  for (int k = threadIdx.x; k < QLORA; k += 256)
    q_norm[(size_t)b * QLORA + k] = q_a[(size_t)b * QLORA + k] * inv * lnw[k];
}

// ---------------------------------------------------------------------------
// q_lat[b,h,c] = sum_d q_nope[b,h,d] * q_absorb[h,d,c]
// ---------------------------------------------------------------------------
__global__ __launch_bounds__(128) void k_qlat(const float* __restrict__ q_full,
                                              const float* __restrict__ Wkv_b,
                                              float* __restrict__ query) {
  const int h = blockIdx.y;
  const int lane = threadIdx.x & 31;
  const int wave = threadIdx.x >> 5;
  const int n0 = blockIdx.x * 64 + wave * 16;
  const float* A = q_full + (size_t)h * QHD;
  const float* B = Wkv_b + (size_t)h * (256 * KVLORA);
#pragma unroll
  for (int mt = 0; mt < 2; ++mt) {
    v8f c = tile_AxB(A, NHEADS * QHD, B, KVLORA, mt * 16, n0, NOPED, lane);
    store_tile(query + (size_t)h * KVDIM, NHEADS * KVDIM, mt * 16, n0, c, lane, SM_SCALE);
  }
}

// ---------------------------------------------------------------------------
// RoPE on q_pe
// ---------------------------------------------------------------------------
__global__ __launch_bounds__(256) void k_rope(const float* __restrict__ q_full,
                                              const int* __restrict__ pos_ids,
                                              float* __restrict__ query) {
  const int tid = blockIdx.x * blockDim.x + threadIdx.x;
  if (tid >= BATCH * NHEADS * 32) return;
  const int i = tid & 31;
  const int h = (tid >> 5) & (NHEADS - 1);
  const int b = tid >> 12;
  const size_t qoff = (size_t)b * (NHEADS * QHD) + (size_t)h * QHD + NOPED;
  const float x = q_full[qoff + 2 * i];
  const float y = q_full[qoff + 2 * i + 1];
  const float pos = (float)pos_ids[b];
  const float inv_freq = powf(10000.0f, -(float)i / 32.0f);
  const float f = pos * inv_freq;
  const float cs = cosf(f), sn = sinf(f);
  float* qd = query + ((size_t)b * NHEADS + h) * KVDIM + KVLORA;
  qd[i]      = (x * cs - y * sn) * SM_SCALE;
  qd[i + 32] = (y * cs + x * sn) * SM_SCALE;
}

// ---------------------------------------------------------------------------
// Flash-decode attention, TDM-staged LDS tiles.
// grid (batch=32, head-group=8), 4 waves, 16 heads, 16 keys/iter.
// Prologue: TDM loads query tile (once) + primes double-buffered KV tiles.
// Wave 0 drives the Tensor Data Mover; tiles gated by s_wait_tensorcnt +
// workgroup barrier.  Scores: waves split K=576 into 144-chunks; both A and
// B fragments come from LDS with a shared padded index.  P@ckv: each wave
// owns 128 latent cols (8 WMMA C-tiles).
// ---------------------------------------------------------------------------
__global__ __launch_bounds__(128) void k_attn(const float* __restrict__ query,
                                              const float* __restrict__ kv,
                                              float* __restrict__ attn_out) {
  const int b = blockIdx.x, hg = blockIdx.y;
  const int lane = threadIdx.x & 31;
  const int wave = threadIdx.x >> 5;
  const int r  = lane & 15;
  const int hh = (lane >> 4) << 1;
  const int h8 = (lane >> 4) * 8;

  __shared__ float kbuf[2 * BUF_DW];   // 74,016 B: double-buffered KV tile
  __shared__ float qsb[BUF_DW];        // 37,008 B: query tile (loaded once)
  __shared__ float S[4][16][16];
  __shared__ float P[16][20];
  __shared__ float mrun[16], lrun[16], fac[16];

  const float* qbase = query + ((size_t)b * NHEADS + (size_t)hg * 16) * KVDIM;
  const float* kvb   = kv + (size_t)b * KVLEN * KVDIM;
  const unsigned kb_off = (unsigned)(size_t)(&kbuf[0]);   // LDS byte offsets
  const unsigned qs_off = (unsigned)(size_t)(&qsb[0]);

  if (threadIdx.x < 16) { mrun[threadIdx.x] = -1e30f; lrun[threadIdx.x] = 0.f; }

  v8f acc[8];
#pragma unroll
  for (int t = 0; t < 8; ++t) acc[t] = (v8f){0.f, 0.f, 0.f, 0.f, 0.f, 0.f, 0.f, 0.f};

#if HAVE_TDM
  if (wave == 0) {                     // query tile + prime both KV buffers
    tdm_load_tile(qs_off, qbase);      // (in-order with KV tile 0)
    tdm_load_tile(kb_off, kvb);
    tdm_load_tile(kb_off + BUF_DW * 4u, kvb + TILE_DW);
  }
#else
  for (int i = threadIdx.x; i < TILE_DW; i += 128)
    qsb[i + (i >> 8)] = qbase[i];
#endif
  __syncthreads();

  // shared padded LDS index base for query A-frag and KV B-frag (phase 1)
  const int Lbase = r * 576 + wave * 144 + hh;

  const int NTILES = KVLEN / 16;                  // 256
  for (int it = 0; it < NTILES; ++it) {
    const int p = it & 1;
    const float* kb = &kbuf[p * BUF_DW];

#if HAVE_TDM
    if (wave == 0) {    // query + tiles <= it resident when <=1 outstanding
      if (it + 2 >= NTILES) __builtin_amdgcn_s_wait_tensorcnt((short)0);
      else                  __builtin_amdgcn_s_wait_tensorcnt((short)1);
    }
#else
    for (int i = threadIdx.x; i < TILE_DW; i += 128)
      kbuf[p * BUF_DW + i + (i >> 8)] = kvb[(size_t)it * TILE_DW + i];
#endif
    __syncthreads();

    // -- phase 1: partial scores from LDS, K chunk [wave*144, +144)
    v8f s = {0.f, 0.f, 0.f, 0.f, 0.f, 0.f, 0.f, 0.f};
#pragma unroll 4
    for (int k = 0; k < 144; k += 4) {
      const int L = Lbase + k;
      const int idx = L + (L >> 8);               // padded LDS layout
      v2f a  = {qsb[idx], qsb[idx + 1]};
      v2f bb = {kb[idx], kb[idx + 1]};
      s = wmma4(a, bb, s);
    }
#pragma unroll
    for (int t = 0; t < 8; ++t) S[wave][t + h8][r] = s[t];
    __syncthreads();

    // -- phase 2: reduce + online softmax (wave 0: 1 row x 8 cols per lane,
    //    halves combined with wave32 shfl_xor)
    if (wave == 0) {
      const int row = r;
      const int nb  = (lane >> 4) * 8;            // col base: 0 or 8
      float m = -1e30f;
      float v[8];
#pragma unroll
      for (int n = 0; n < 8; ++n) {
        float t = S[0][row][nb + n] + S[1][row][nb + n]
                + S[2][row][nb + n] + S[3][row][nb + n];
        v[n] = t;
        m = fmaxf(m, t);
      }
      m = fmaxf(m, __shfl_xor(m, 16, 32));        // tile max over 16 cols
      const float mold = mrun[row];               // lockstep read before write
      m = fmaxf(m, mold);
      float l = 0.f;
#pragma unroll
      for (int n = 0; n < 8; ++n) {
        float pv = __expf(v[n] - m);
        P[row][nb + n] = pv;
        l += pv;
      }
      l += __shfl_xor(l, 16, 32);                 // row sum over 16 cols
      if (lane < 16) {
        const float f = __expf(mold - m);
        fac[row]  = f;
        mrun[row] = m;
        lrun[row] = lrun[row] * f + l;
      }
    }
    __syncthreads();

    // -- phase 3: rescale accumulators, acc += P(16x16) @ ckv(16x512) from LDS
    float fr[8];
#pragma unroll
    for (int t = 0; t < 8; ++t) fr[t] = fac[t + h8];

    const float* pb = &P[r][hh];
#pragma unroll
    for (int nt = 0; nt < 8; ++nt) {
      const int n0 = wave * 128 + nt * 16;          // latent col < 512
      const int Lb = hh * 576 + n0 + r;
      v8f c = acc[nt];
#pragma unroll
      for (int t = 0; t < 8; ++t) c[t] *= fr[t];
#pragma unroll
      for (int kk = 0; kk < 16; kk += 4) {
        v2f a = {pb[kk], pb[kk + 1]};
        const int L1 = Lb + kk * 576;
        const int L2 = L1 + 576;
        v2f bb = {kb[L1 + (L1 >> 8)], kb[L2 + (L2 >> 8)]};
        c = wmma4(a, bb, c);
      }
      acc[nt] = c;
    }
    __syncthreads();                  // everyone done with buffer p

#if HAVE_TDM
    if (wave == 0 && it + 2 < NTILES)  // refill buffer p with tile it+2
      tdm_load_tile(kb_off + (unsigned)p * (BUF_DW * 4u),
                    kvb + (size_t)(it + 2) * TILE_DW);
#endif
  }

  // -- epilogue: normalize, write attn_out[b, h, 0:512]
  float linv[8];
#pragma unroll
  for (int t = 0; t < 8; ++t) linv[t] = 1.0f / lrun[t + h8];

  float* ob = attn_out + ((size_t)b * NHEADS + (size_t)hg * 16) * KVLORA;
#pragma unroll
  for (int nt = 0; nt < 8; ++nt) {
    const int n0 = wave * 128 + nt * 16;
#pragma unroll
    for (int t = 0; t < 8; ++t)
      ob[(size_t)(t + h8) * KVLORA + n0 + r] = acc[nt][t] * linv[t];
  }
}

// ---------------------------------------------------------------------------
// o[b,h,v] = sum_c attn[b,h,c] * out_absorb[h,v,c]
// ---------------------------------------------------------------------------
__global__ __launch_bounds__(128) void k_oabs(const float* __restrict__ attn_o,
                                              const float* __restrict__ Wkv_b,
                                              float* __restrict__ o_heads) {
  const int h = blockIdx.y;
  const int lane = threadIdx.x & 31;
  const int wave = threadIdx.x >> 5;
  const int n0 = blockIdx.x * 64 + wave * 16;
  const float* A = attn_o + (size_t)h * KVLORA;
  const float* W = Wkv_b + (size_t)h * (256 * KVLORA) + (size_t)NOPED * KVLORA;
#pragma unroll
  for (int mt = 0; mt < 2; ++mt) {
    v8f c = tile_AxWt(A, NHEADS * KVLORA, W, KVLORA, mt * 16, n0, KVLORA, lane);
    store_tile(o_heads + (size_t)h * VD, NHEADS * VD, mt * 16, n0, c, lane, 1.0f);
  }
}

// ---------------------------------------------------------------------------
// Host launcher
// ---------------------------------------------------------------------------
extern "C" void kernel_launch(void* const* d_in, const int* in_sizes, int n_in,
                              void* d_out, int out_size, void* d_ws, size_t ws_size,
                              hipStream_t stream) {
  (void)in_sizes; (void)n_in; (void)out_size; (void)ws_size;
  const float* hidden = (const float*)d_in[0];
  const int*   pos    = (const int*)d_in[1];
  const float* kv     = (const float*)d_in[2];
  const float* Wq_a   = (const float*)d_in[3];
  const float* lnw    = (const float*)d_in[4];
  const float* Wq_b   = (const float*)d_in[5];
  const float* Wkv_b  = (const float*)d_in[6];
  const float* Wo     = (const float*)d_in[7];
  float* out = (float*)d_out;

  float* ws      = (float*)d_ws;
  float* q_a     = ws;
  float* q_norm  = q_a + (size_t)BATCH * QLORA;
  float* q_full  = q_norm + (size_t)BATCH * QLORA;
  float* query   = q_full + (size_t)BATCH * NHEADS * QHD;
  float* attn_o  = query + (size_t)BATCH * NHEADS * KVDIM;
  float* o_heads = q_full;   // reuse: q_full dead after rope

  k_gemm32_AxWt<<<QLORA / 64, 128, 0, stream>>>(hidden, HIDDEN, Wq_a, HIDDEN,
                                                q_a, QLORA, HIDDEN);
  k_rms<<<BATCH, 256, 0, stream>>>(q_a, lnw, q_norm);
  k_gemm32_AxWt<<<(NHEADS * QHD) / 64, 128, 0, stream>>>(q_norm, QLORA, Wq_b, QLORA,
                                                         q_full, NHEADS * QHD, QLORA);
  {
    dim3 g(KVLORA / 64, NHEADS);
    k_qlat<<<g, 128, 0, stream>>>(q_full, Wkv_b, query);
  }
  k_rope<<<(BATCH * NHEADS * 32) / 256, 256, 0, stream>>>(q_full, pos, query);
  {
    dim3 g(BATCH, NHEADS / 16);
    k_attn<<<g, 128, 0, stream>>>(query, kv, attn_o);
  }
  {
    dim3 g(VD / 64, NHEADS);
    k_oabs<<<g, 128, 0, stream>>>(attn_o, Wkv_b, o_heads);
  }
  k_gemm32_AxWt<<<HIDDEN / 64, 128, 0, stream>>>(o_heads, NHEADS * VD, Wo, NHEADS * VD,
                                                 out, HIDDEN, NHEADS * VD);
}